// Aggregator_29119878266988
// MI455X (gfx1250) — compile-verified
//
#include <hip/hip_runtime.h>

#define THREADS 256
#define EDGES_PER_TILE 16   // 256 threads x 16B async chunks = 4KB = 16 edges x 256B
#define N_FEAT 64

__global__ void zero_f4(float4* __restrict__ out, int n4) {
    int i = blockIdx.x * blockDim.x + threadIdx.x;
    if (i < n4) out[i] = make_float4(0.f, 0.f, 0.f, 0.f);
}

__global__ __launch_bounds__(THREADS) void segsum_async(
    const float* __restrict__ edges,
    const int*   __restrict__ receivers,
    float*       __restrict__ out,
    int n_edges)
{
    __shared__ float4 stage[THREADS];   // 4KB staging buffer, starts at LDS offset 0

    const unsigned tid   = threadIdx.x;
    const unsigned tile  = blockIdx.x;
    const unsigned edge  = tile * EDGES_PER_TILE + (tid >> 4);  // 16 threads per edge row
    const unsigned chunk = tid & 15u;                           // which float4 of the row
    const bool valid = (edge < (unsigned)n_edges);

    // Global byte offset of this thread's 16B chunk within edges[] (max 256MB, fits u32).
    const unsigned goff = valid ? (edge * (N_FEAT * 4u) + chunk * 16u) : 0u;
    // LDS byte address of this thread's staging slot (low 32 bits of the flat
    // shared-aperture address are the LDS offset on gfx1250).
    const unsigned lds_addr = (unsigned)(size_t)(&stage[tid]);

    // CDNA5 async DMA: global -> LDS, 16B/lane, tracked with ASYNCcnt.
    // NT hint: the 256MB edge stream is touched once -- keep it from evicting
    // the atomically-hot 25.6MB output region out of the 192MB L2.
    asm volatile("global_load_async_to_lds_b128 %0, %1, %2 th:TH_LOAD_NT"
                 :
                 : "v"(lds_addr), "v"(goff), "s"(edges)
                 : "memory");

    // Overlap the receiver-index load with the in-flight DMA.
    const int recv = valid ? receivers[edge] : 0;

    // Per-wave wait: each thread reads back exactly the chunk it staged,
    // so no workgroup barrier is needed.
    asm volatile("s_wait_asynccnt 0x0" ::: "memory");

    const float4 v = stage[tid];

    if (valid) {
        // Byte offset into out[] for this chunk (max 25.6MB, fits u32).
        const unsigned obase = (unsigned)recv * (N_FEAT * 4u) + chunk * 16u;
        // No-return f32 atomic adds resolve in the global L2 (output is
        // L2-resident); device scope for cross-WGP coherence.
        asm volatile("global_atomic_add_f32 %0, %1, %2 scope:SCOPE_DEV"
                     :: "v"(obase), "v"(v.x), "s"(out) : "memory");
        asm volatile("global_atomic_add_f32 %0, %1, %2 offset:4 scope:SCOPE_DEV"
                     :: "v"(obase), "v"(v.y), "s"(out) : "memory");
        asm volatile("global_atomic_add_f32 %0, %1, %2 offset:8 scope:SCOPE_DEV"
                     :: "v"(obase), "v"(v.z), "s"(out) : "memory");
        asm volatile("global_atomic_add_f32 %0, %1, %2 offset:12 scope:SCOPE_DEV"
                     :: "v"(obase), "v"(v.w), "s"(out) : "memory");
    }
}

extern "C" void kernel_launch(void* const* d_in, const int* in_sizes, int n_in,
                              void* d_out, int out_size, void* d_ws, size_t ws_size,
                              hipStream_t stream) {
    const float* edges     = (const float*)d_in[0];   // [n_edges, 64] f32
    const int*   receivers = (const int*)  d_in[2];   // [n_edges] i32
    float*       out       = (float*)d_out;           // [n_nodes, 64] f32

    const int n_edges = in_sizes[0] / N_FEAT;

    // 1) Zero the output (harness poisons d_out; segments with no edges must be 0).
    const int n4 = out_size / 4;
    zero_f4<<<(n4 + THREADS - 1) / THREADS, THREADS, 0, stream>>>((float4*)out, n4);

    // 2) Async-staged scatter-add: one 16-edge tile per 256-thread block.
    const int n_tiles = (n_edges + EDGES_PER_TILE - 1) / EDGES_PER_TILE;
    segsum_async<<<n_tiles, THREADS, 0, stream>>>(edges, receivers, out, n_edges);
}